// Model_63170378989866
// MI455X (gfx1250) — compile-verified
//
#include <hip/hip_runtime.h>
#include <hip/hip_bf16.h>
#include <math.h>

// ---------------- constants (match reference) ----------------
#define B_   256
#define N_   100
#define E_   128
#define M_   8
#define HD_  16
#define R_   (B_ * N_)          // 25600 tokens
#define TSTEPS (2 * N_)         // 200 decode steps
#define CLIP_ 10.0f
#define INV_SQRT_HD 0.25f       // 1/sqrt(16)

typedef float v2f  __attribute__((ext_vector_type(2)));
typedef float v8f  __attribute__((ext_vector_type(8)));
typedef unsigned int u32x4 __attribute__((ext_vector_type(4)));
typedef int  i32x4 __attribute__((ext_vector_type(4)));
typedef int  i32x8 __attribute__((ext_vector_type(8)));

// =====================================================================
// TDM helper: DMA a 32x128 f32 tile (row-major, stride 128 floats) from
// global memory into LDS.  D# built per CDNA5 ISA §8.3/8.4:
//   group0: count=1 | lds_addr | global_addr(57b) | type=2
//   group1: data_size=4B, tensor_dim0=128, tensor_dim1=32,
//           tile_dim0=128, tile_dim1=32, tensor_dim0_stride=128
//   groups 2/3 (+extra group): zero (<=2D tensor)
// Tracked by TENSORcnt.  (6-arg builtin form on this toolchain.)
// =====================================================================
__device__ __forceinline__ void tdm_load_chunk(const float* gsrc, void* ldsDst) {
  const unsigned long long ga = (unsigned long long)(size_t)gsrc;
  const unsigned lds = (unsigned)(size_t)ldsDst;   // low 32 bits = LDS offset
  u32x4 g0;
  g0.x = 1u;                                        // count=1 (valid), user mode
  g0.y = lds;                                       // lds_addr [63:32]
  g0.z = (unsigned)(ga & 0xffffffffu);              // global_addr [95:64]
  g0.w = (unsigned)((ga >> 32) & 0x01ffffffu)       // global_addr [120:96]
       | (2u << 30);                                // type=2 ("image") [127:126]
  i32x8 g1;
  g1[0] = 0x00020000;          // workgroup_mask=0, data_size=2 (4 bytes)
  g1[1] = (int)(128u << 16);   // tensor_dim0[15:0]=128 at bits [63:48]
  g1[2] = (int)(32u << 16);    // tensor_dim1[15:0]=32  at bits [111:80]
  g1[3] = (int)(128u << 16);   // tile_dim0=128 at bits [127:112]
  g1[4] = 32;                  // tile_dim1=32 at bits [143:128]
  g1[5] = 128;                 // tensor_dim0_stride low32 = 128 elements
  g1[6] = 0;
  g1[7] = 0;
  const i32x4 gz4 = {0, 0, 0, 0};
  const i32x8 gz8 = {0, 0, 0, 0, 0, 0, 0, 0};
  __builtin_amdgcn_tensor_load_to_lds(g0, g1, gz4, gz4, gz8, 0);
}

// =====================================================================
// FP32 GEMM via V_WMMA_F32_16X16X4_F32 with TDM-staged weights.
//   C(25600 x 128) = A @ W(128x128) + bias [+ residual]
// Block = 256 thr = 8 waves = 128 rows of A.  W streamed into LDS in four
// 32x128-f32 chunks, double buffered (issue next / wait previous / barrier).
// Each wave keeps a 16x128 strip: one A fragment feeds 8 WMMAs.
//   A frag (16x4): lane L<16 row M=L, K=kk+{0,1}; lanes 16-31 K=kk+{2,3}
//   B frag (4x16): lane holds column N=(lane&15), same K split
//   C/D: vgpr r -> M = r + (lane>=16 ? 8 : 0), N = lane&15
// =====================================================================
template <bool HAS_RES>
__global__ __launch_bounds__(256)
void gemm_e128_wmma(const float* __restrict__ A, const float* __restrict__ W,
                    const float* __restrict__ bias, const float* __restrict__ residual,
                    float* __restrict__ C) {
  __shared__ float wbuf[2][32 * E_];        // 2 x 16 KB

  const int wid  = threadIdx.x >> 5;
  const int lane = threadIdx.x & 31;
  const int rowBase = (blockIdx.x * 8 + wid) * 16;
  const int row     = rowBase + (lane & 15);
  const int khalf   = (lane >> 4) * 2;      // 0 or 2
  const float* Arow = A + (size_t)row * E_;

  if (wid == 0) tdm_load_chunk(W, &wbuf[0][0]);

  v8f acc[8] = {};
  for (int chunk = 0; chunk < 4; ++chunk) {
    if (wid == 0) {
      if (chunk + 1 < 4) {
        tdm_load_chunk(W + (size_t)(chunk + 1) * 32 * E_, &wbuf[(chunk + 1) & 1][0]);
        __builtin_amdgcn_s_wait_tensorcnt(1);   // previous chunk complete
      } else {
        __builtin_amdgcn_s_wait_tensorcnt(0);
      }
    }
    __syncthreads();                            // chunk data visible to all waves

    const float* wb = &wbuf[chunk & 1][0];
    const int kbase = chunk * 32;
    #pragma unroll
    for (int kkl = 0; kkl < 32; kkl += 4) {
      v2f a;
      a.x = Arow[kbase + kkl + khalf];
      a.y = Arow[kbase + kkl + khalf + 1];
      #pragma unroll
      for (int tn = 0; tn < 8; ++tn) {
        v2f b;
        b.x = wb[(kkl + khalf)     * E_ + tn * 16 + (lane & 15)];
        b.y = wb[(kkl + khalf + 1) * E_ + tn * 16 + (lane & 15)];
        acc[tn] = __builtin_amdgcn_wmma_f32_16x16x4_f32(
            false, a, false, b, (short)0, acc[tn], false, false);
      }
    }
    __syncthreads();                            // all done reading before reuse
  }

  const int rbase = rowBase + ((lane >> 4) ? 8 : 0);
  #pragma unroll
  for (int tn = 0; tn < 8; ++tn) {
    const int col = tn * 16 + (lane & 15);
    const float bval = bias[col];
    #pragma unroll
    for (int r = 0; r < 8; ++r) {
      float v = acc[tn][r] + bval;
      if (HAS_RES) v += residual[(size_t)(rbase + r) * E_ + col];
      C[(size_t)(rbase + r) * E_ + col] = v;
    }
  }
}

// =====================================================================
// Haversine distance matrix: dis[b][i][j]
// =====================================================================
__global__ void haversine_kernel(const float* __restrict__ s, float* __restrict__ dis) {
  const long long idx = (long long)blockIdx.x * blockDim.x + threadIdx.x;
  const long long total = (long long)B_ * N_ * N_;
  if (idx >= total) return;
  const int b = (int)(idx / (N_ * N_));
  const int r = (int)(idx % (N_ * N_));
  const int i = r / N_, j = r % N_;
  const float D2R = 0.017453292519943295f;
  const float lat1 = s[((size_t)(b * N_ + i)) * 2 + 0] * D2R;
  const float lon1 = s[((size_t)(b * N_ + i)) * 2 + 1] * D2R;
  const float lat2 = s[((size_t)(b * N_ + j)) * 2 + 0] * D2R;
  const float lon2 = s[((size_t)(b * N_ + j)) * 2 + 1] * D2R;
  const float sdlat = sinf((lat2 - lat1) * 0.5f);
  const float sdlon = sinf((lon2 - lon1) * 0.5f);
  float a = sdlat * sdlat + cosf(lat1) * cosf(lat2) * sdlon * sdlon;
  a = fminf(fmaxf(a, 0.0f), 1.0f);
  dis[idx] = 6367.0f * 2.0f * asinf(sqrtf(a));
}

// =====================================================================
// Embeddings: sd = emb1([s,d]) (row 0 -> emb_p(s)), tw = emb2([start,stop])
// grid = B*N blocks, 128 threads
// =====================================================================
__global__ void embed_kernel(const float* __restrict__ s, const int* __restrict__ d,
                             const float* __restrict__ startT, const float* __restrict__ stopT,
                             const float* __restrict__ W1, const float* __restrict__ b1,
                             const float* __restrict__ W2, const float* __restrict__ b2,
                             const float* __restrict__ Wp, const float* __restrict__ bp,
                             float* __restrict__ sd, float* __restrict__ tw) {
  const int bn = blockIdx.x;
  const int e  = threadIdx.x;
  const int n  = bn % N_;
  const float s0 = s[(size_t)bn * 2 + 0];
  const float s1 = s[(size_t)bn * 2 + 1];
  float sdv;
  if (n == 0) {
    sdv = s0 * Wp[0 * E_ + e] + s1 * Wp[1 * E_ + e] + bp[e];
  } else {
    const float dv = (float)d[bn];
    sdv = s0 * W1[0 * E_ + e] + s1 * W1[1 * E_ + e] + dv * W1[2 * E_ + e] + b1[e];
  }
  sd[(size_t)bn * E_ + e] = sdv;
  const float t0 = startT[bn], t1 = stopT[bn];
  tw[(size_t)bn * E_ + e] = t0 * W2[0 * E_ + e] + t1 * W2[1 * E_ + e] + b2[e];
}

// =====================================================================
// Encoder attention: one block per (b, h). Scores in LDS (100x104 padded).
// Writes output in-place over the q buffer (each block owns its slice).
// =====================================================================
__global__ void attn_kernel(float* __restrict__ q, const float* __restrict__ k,
                            const float* __restrict__ v) {
  const int b = blockIdx.x / M_;
  const int h = blockIdx.x % M_;
  const int tid = threadIdx.x;          // blockDim = 256

  __shared__ float qs[N_ * HD_];        // 6.4 KB (reused for v later)
  __shared__ float ks[N_ * HD_];        // 6.4 KB
  __shared__ float S[N_ * 104];         // 41.6 KB (padded rows)

  for (int idx = tid; idx < N_ * HD_; idx += 256) {
    const int n = idx >> 4, dch = idx & 15;
    const size_t base = ((size_t)(b * N_ + n)) * E_ + h * HD_;
    qs[idx] = q[base + dch];
    ks[idx] = k[base + dch];
  }
  __syncthreads();

  for (int idx = tid; idx < N_ * N_; idx += 256) {
    const int i = idx / N_, j = idx % N_;
    float acc = 0.0f;
    #pragma unroll
    for (int dch = 0; dch < HD_; ++dch) acc += qs[i * HD_ + dch] * ks[j * HD_ + dch];
    S[i * 104 + j] = acc * INV_SQRT_HD;
  }
  __syncthreads();

  if (tid < N_) {
    const int i = tid;
    float mx = -INFINITY;
    for (int j = 0; j < N_; ++j) mx = fmaxf(mx, S[i * 104 + j]);
    float sum = 0.0f;
    for (int j = 0; j < N_; ++j) { const float e = expf(S[i * 104 + j] - mx); S[i * 104 + j] = e; sum += e; }
    const float r = 1.0f / sum;
    for (int j = 0; j < N_; ++j) S[i * 104 + j] *= r;
  }
  __syncthreads();

  for (int idx = tid; idx < N_ * HD_; idx += 256) {
    const int n = idx >> 4, dch = idx & 15;
    qs[idx] = v[((size_t)(b * N_ + n)) * E_ + h * HD_ + dch];
  }
  __syncthreads();

  for (int idx = tid; idx < N_ * HD_; idx += 256) {
    const int i = idx >> 4, dch = idx & 15;
    float acc = 0.0f;
    for (int j = 0; j < N_; ++j) acc += S[i * 104 + j] * qs[j * HD_ + dch];
    q[((size_t)(b * N_ + i)) * E_ + h * HD_ + dch] = acc;   // in-place output
  }
}

// =====================================================================
// x = sd + tw (in-place into sd), avg = mean over N.  grid = B, block = 128
// =====================================================================
__global__ void add_mean_kernel(float* __restrict__ sd, const float* __restrict__ tw,
                                float* __restrict__ avg) {
  const int b = blockIdx.x, e = threadIdx.x;
  float acc = 0.0f;
  for (int n = 0; n < N_; ++n) {
    const size_t o = ((size_t)(b * N_ + n)) * E_ + e;
    const float vv = sd[o] + tw[o];
    sd[o] = vv;
    acc += vv;
  }
  avg[(size_t)b * E_ + e] = acc * (1.0f / N_);
}

// =====================================================================
// Decoder state init.  grid = B, block = 128
// =====================================================================
__global__ void init_state(const int* __restrict__ d, const int* __restrict__ lptr,
                           int* idx_cur, int* tag, int* rest, int* dd, int* dd_sum,
                           float* cur_time, float* total_cost, int* done_eff) {
  const int b = blockIdx.x, tid = threadIdx.x;
  __shared__ int ssum;
  if (tid == 0) ssum = 0;
  __syncthreads();
  int local = 0;
  for (int n = tid; n < N_; n += 128) {
    const int dv = d[b * N_ + n];
    dd[b * N_ + n] = dv;
    tag[b * N_ + n] = 1;
    local += dv;
  }
  atomicAdd(&ssum, local);
  __syncthreads();
  if (tid == 0) {
    dd_sum[b]     = ssum;
    idx_cur[b]    = 0;
    rest[b]       = lptr[0];
    cur_time[b]   = 0.0f;
    total_cost[b] = 0.0f;
    if (b == 0) done_eff[0] = 0;
  }
}

// =====================================================================
// Per-step reduction: finished[b] = (dd_sum[b]==0); eff = done | all(finished)
// =====================================================================
__global__ void decode_reduce(const int* __restrict__ dd_sum, int* __restrict__ finished,
                              int* __restrict__ done_eff) {
  __shared__ int allfin;
  if (threadIdx.x == 0) allfin = 1;
  __syncthreads();
  const int b = threadIdx.x;
  const int fin = (dd_sum[b] == 0) ? 1 : 0;
  finished[b] = fin;
  if (!fin) atomicAnd(&allfin, 0);
  __syncthreads();
  if (threadIdx.x == 0) {
    const int eff = done_eff[0] | allfin;
    done_eff[0] = eff;
  }
}

// =====================================================================
// One fused decode step for all batch elements. grid = B, block = 128.
// =====================================================================
__global__ void decode_step(int t,
                            const float* __restrict__ avg, const float* __restrict__ x,
                            const float* __restrict__ keyv, const float* __restrict__ value,
                            const float* __restrict__ k2, const float* __restrict__ dis,
                            const float* __restrict__ startT, const float* __restrict__ stopT,
                            const float* __restrict__ wqW, const float* __restrict__ wqb,
                            const float* __restrict__ wW,  const float* __restrict__ wb,
                            const float* __restrict__ qW,  const float* __restrict__ qb,
                            const int* __restrict__ lptr,
                            int* idx_cur, int* tag, int* rest, int* dd, int* dd_sum,
                            float* cur_time, float* total_cost,
                            const int* __restrict__ finished_arr, const int* __restrict__ done_eff,
                            float* __restrict__ out_seq, float* __restrict__ out_pro) {
  const int b   = blockIdx.x;
  const int tid = threadIdx.x;
  const int eff      = done_eff[0];
  const int finished = finished_arr[b];
  const int index    = idx_cur[b];
  const int restv    = rest[b];

  __shared__ float graph[2 * E_ + 1];
  __shared__ float query[E_];
  __shared__ float att[N_ * M_];
  __shared__ float glimpse[E_];
  __shared__ float hidden[E_];
  __shared__ float q2s[E_];
  __shared__ float logits[N_];
  __shared__ int   maskS[N_];

  graph[tid]       = avg[(size_t)b * E_ + tid];
  graph[E_ + tid]  = x[((size_t)(b * N_ + index)) * E_ + tid];
  if (tid == 0) graph[2 * E_] = (float)restv;

  for (int n = tid; n < N_; n += 128) {
    const int tagv = (n == index) ? 0 : tag[b * N_ + n];
    int m = (!tagv) || (dd[b * N_ + n] > restv);
    if (n == 0) {
      const int depot_open = finished || (index != 0);
      m = m && !depot_open;
    }
    maskS[n] = m;
  }
  __syncthreads();

  {
    float acc = wqb[tid];
    for (int f = 0; f < 2 * E_ + 1; ++f) acc += graph[f] * wqW[(size_t)f * E_ + tid];
    query[tid] = acc;
  }
  __syncthreads();

  for (int idx = tid; idx < N_ * M_; idx += 128) {
    const int n = idx >> 3, h = idx & 7;
    const float* kv = &keyv[((size_t)(b * N_ + n)) * E_ + h * HD_];
    float acc = 0.0f;
    #pragma unroll
    for (int dch = 0; dch < HD_; ++dch) acc += query[h * HD_ + dch] * kv[dch];
    att[n * M_ + h] = acc * INV_SQRT_HD;
  }
  __syncthreads();

  if (tid < M_) {
    const int h = tid;
    float mx = -INFINITY;
    for (int n = 0; n < N_; ++n) if (!maskS[n]) mx = fmaxf(mx, att[n * M_ + h]);
    float sum = 0.0f;
    for (int n = 0; n < N_; ++n) {
      const float e = maskS[n] ? 0.0f : expf(att[n * M_ + h] - mx);
      att[n * M_ + h] = e; sum += e;
    }
    const float r = (sum > 0.0f) ? 1.0f / sum : 0.0f;
    for (int n = 0; n < N_; ++n) att[n * M_ + h] *= r;
  }
  __syncthreads();

  {
    const int h = tid >> 4;
    float acc = 0.0f;
    for (int n = 0; n < N_; ++n) acc += att[n * M_ + h] * value[((size_t)(b * N_ + n)) * E_ + tid];
    glimpse[tid] = acc;
  }
  __syncthreads();

  {
    float acc = wb[tid];
    for (int f = 0; f < E_; ++f) acc += glimpse[f] * wW[(size_t)f * E_ + tid];
    hidden[tid] = acc;
  }
  __syncthreads();
  {
    float acc = qb[tid];
    for (int f = 0; f < E_; ++f) acc += hidden[f] * qW[(size_t)f * E_ + tid];
    q2s[tid] = acc;
  }
  __syncthreads();

  for (int n = tid; n < N_; n += 128) {
    const float* kk = &k2[((size_t)(b * N_ + n)) * E_];
    float acc = 0.0f;
    for (int e = 0; e < E_; ++e) acc += q2s[e] * kk[e];
    logits[n] = tanhf(acc * INV_SQRT_HD) * CLIP_;
  }
  __syncthreads();

  if (tid == 0) {
    float mx = -INFINITY;
    for (int n = 0; n < N_; ++n) if (!maskS[n]) mx = fmaxf(mx, logits[n]);
    float sum = 0.0f;
    for (int n = 0; n < N_; ++n) {
      const float e = maskS[n] ? 0.0f : expf(logits[n] - mx);
      logits[n] = e; sum += e;
    }
    const float r = (sum > 0.0f) ? 1.0f / sum : 0.0f;
    int amax = 0; float best = -1.0f;
    for (int n = 0; n < N_; ++n) {
      const float pv = logits[n] * r;
      logits[n] = pv;
      if (pv > best) { best = pv; amax = n; }
    }
    const int indexx  = finished ? 0 : amax;
    const float pro_i = finished ? 1.0f : logits[indexx];
    const int sel = dd[b * N_ + indexx];
    int rest2 = restv - sel;
    const float step_dist = dis[((size_t)(b * N_ + index)) * N_ + indexx];
    float tc = total_cost[b] + step_dist;
    float ct = cur_time[b] + step_dist / (10.0f / 3600.0f);
    const float st = startT[b * N_ + indexx];
    const float sp = stopT[b * N_ + indexx];
    if (st > ct) tc += floorf((st - ct) * 0.2f);
    if (sp < ct) tc += floorf(ct - sp);
    ct += (float)sel * 300.0f;
    if (indexx == 0) rest2 = lptr[0];

    out_seq[(size_t)b * TSTEPS + t] = eff ? 0.0f : (float)indexx;
    out_pro[(size_t)b * TSTEPS + t] = eff ? 1.0f : pro_i;

    if (!eff) {
      idx_cur[b] = indexx;
      tag[b * N_ + index] = 0;
      rest[b] = rest2;
      dd[b * N_ + indexx] = 0;
      dd_sum[b] -= sel;
      cur_time[b] = ct;
      total_cost[b] = tc;
    }
  }
}

__global__ void finalize_cost(const float* __restrict__ total_cost, float* __restrict__ out) {
  const int b = threadIdx.x;
  out[b] = total_cost[b];
}

// =====================================================================
// Host launcher
// =====================================================================
extern "C" void kernel_launch(void* const* d_in, const int* in_sizes, int n_in,
                              void* d_out, int out_size, void* d_ws, size_t ws_size,
                              hipStream_t stream) {
  const float* s      = (const float*)d_in[0];
  const int*   d      = (const int*)  d_in[1];
  const float* startT = (const float*)d_in[2];
  const float* stopT  = (const float*)d_in[3];
  const float* emb1W = (const float*)d_in[4];  const float* emb1b = (const float*)d_in[5];
  const float* emb2W = (const float*)d_in[6];  const float* emb2b = (const float*)d_in[7];
  const float* embpW = (const float*)d_in[8];  const float* embpb = (const float*)d_in[9];
  const float* dwqW = (const float*)d_in[58]; const float* dwqb = (const float*)d_in[59];
  const float* dwkW = (const float*)d_in[60]; const float* dwkb = (const float*)d_in[61];
  const float* dwvW = (const float*)d_in[62]; const float* dwvb = (const float*)d_in[63];
  const float* dwW  = (const float*)d_in[64]; const float* dwb  = (const float*)d_in[65];
  const float* dqW  = (const float*)d_in[66]; const float* dqb  = (const float*)d_in[67];
  const float* dkW  = (const float*)d_in[68]; const float* dkb  = (const float*)d_in[69];
  const int*   lptr = (const int*)d_in[70];
  (void)in_sizes; (void)n_in; (void)out_size; (void)ws_size;

  char* ws = (char*)d_ws;
  size_t off = 0;
  auto alloc = [&](size_t bytes) { void* p = ws + off; off += (bytes + 255) & ~(size_t)255; return p; };
  float* dis = (float*)alloc((size_t)B_ * N_ * N_ * 4);
  float* sd  = (float*)alloc((size_t)R_ * E_ * 4);
  float* tw  = (float*)alloc((size_t)R_ * E_ * 4);
  float* bq  = (float*)alloc((size_t)R_ * E_ * 4);   // later: keyv
  float* bk  = (float*)alloc((size_t)R_ * E_ * 4);   // later: value
  float* bv  = (float*)alloc((size_t)R_ * E_ * 4);   // later: k2
  float* avg = (float*)alloc((size_t)B_ * E_ * 4);
  int* idx_cur = (int*)alloc(B_ * 4);
  int* rest    = (int*)alloc(B_ * 4);
  int* dd      = (int*)alloc((size_t)R_ * 4);
  int* dd_sum  = (int*)alloc(B_ * 4);
  int* tag     = (int*)alloc((size_t)R_ * 4);
  int* finished= (int*)alloc(B_ * 4);
  int* done_eff= (int*)alloc(2 * 4);
  float* cur_time   = (float*)alloc(B_ * 4);
  float* total_cost = (float*)alloc(B_ * 4);

  float* out_seq = (float*)d_out;
  float* out_pro = (float*)d_out + (size_t)B_ * TSTEPS;
  float* out_tc  = out_pro + (size_t)B_ * TSTEPS;

  {
    const long long total = (long long)B_ * N_ * N_;
    haversine_kernel<<<(unsigned)((total + 255) / 256), 256, 0, stream>>>(s, dis);
  }
  embed_kernel<<<R_, E_, 0, stream>>>(s, d, startT, stopT,
                                      emb1W, emb1b, emb2W, emb2b, embpW, embpb, sd, tw);

  const int gemmBlocks = R_ / 128;            // 8 waves x 16 rows per block
  for (int layer = 0; layer < 6; ++layer) {
    float* X = (layer < 3) ? sd : tw;
    const int base = 10 + 8 * layer;
    const float* wqW = (const float*)d_in[base + 0]; const float* wqb = (const float*)d_in[base + 1];
    const float* wkW = (const float*)d_in[base + 2]; const float* wkb = (const float*)d_in[base + 3];
    const float* wvW = (const float*)d_in[base + 4]; const float* wvb = (const float*)d_in[base + 5];
    const float* woW = (const float*)d_in[base + 6]; const float* wob = (const float*)d_in[base + 7];

    gemm_e128_wmma<false><<<gemmBlocks, 256, 0, stream>>>(X, wqW, wqb, nullptr, bq);
    gemm_e128_wmma<false><<<gemmBlocks, 256, 0, stream>>>(X, wkW, wkb, nullptr, bk);
    gemm_e128_wmma<false><<<gemmBlocks, 256, 0, stream>>>(X, wvW, wvb, nullptr, bv);
    attn_kernel<<<B_ * M_, 256, 0, stream>>>(bq, bk, bv);
    gemm_e128_wmma<true><<<gemmBlocks, 256, 0, stream>>>(bq, woW, wob, X, X);
  }

  add_mean_kernel<<<B_, E_, 0, stream>>>(sd, tw, avg);
  float* x = sd;
  gemm_e128_wmma<false><<<gemmBlocks, 256, 0, stream>>>(x, dwkW, dwkb, nullptr, bq);  // keyv
  gemm_e128_wmma<false><<<gemmBlocks, 256, 0, stream>>>(x, dwvW, dwvb, nullptr, bk);  // value
  gemm_e128_wmma<false><<<gemmBlocks, 256, 0, stream>>>(x, dkW,  dkb,  nullptr, bv);  // k2

  init_state<<<B_, 128, 0, stream>>>(d, lptr, idx_cur, tag, rest, dd, dd_sum,
                                     cur_time, total_cost, done_eff);
  for (int t = 0; t < TSTEPS; ++t) {
    decode_reduce<<<1, 256, 0, stream>>>(dd_sum, finished, done_eff);
    decode_step<<<B_, 128, 0, stream>>>(t, avg, x, bq, bk, bv, dis, startT, stopT,
                                        dwqW, dwqb, dwW, dwb, dqW, dqb, lptr,
                                        idx_cur, tag, rest, dd, dd_sum,
                                        cur_time, total_cost, finished, done_eff,
                                        out_seq, out_pro);
  }
  finalize_cost<<<1, B_, 0, stream>>>(total_cost, out_tc);
}